// EdgeLayer_82824149336364
// MI455X (gfx1250) — compile-verified
//
#include <hip/hip_runtime.h>

typedef __attribute__((ext_vector_type(2))) float v2f;
typedef __attribute__((ext_vector_type(8))) float v8f;

#define DIM 128  // IN_DIM == HIDDEN == 128

// ---------------------------------------------------------------------------
// 1) deg[i] = 1.0f  (self-loop contribution to in-degree)
// ---------------------------------------------------------------------------
__global__ void deg_init_kernel(float* __restrict__ deg, int n) {
    int i = blockIdx.x * blockDim.x + threadIdx.x;
    if (i < n) deg[i] = 1.0f;
}

// ---------------------------------------------------------------------------
// 2) deg[dst[e]] += 1.0f for every edge (float atomic, L2-resident array)
// ---------------------------------------------------------------------------
__global__ void deg_accum_kernel(const long long* __restrict__ dst,
                                 float* __restrict__ deg, int e) {
    int i = blockIdx.x * blockDim.x + threadIdx.x;
    if (i < e) atomicAdd(&deg[(int)dst[i]], 1.0f);
}

// ---------------------------------------------------------------------------
// 3) deg -> deg^{-1/2} in place (deg >= 1 always, no zero check needed)
// ---------------------------------------------------------------------------
__global__ void dinv_kernel(float* __restrict__ deg, int n) {
    int i = blockIdx.x * blockDim.x + threadIdx.x;
    if (i < n) deg[i] = rsqrtf(deg[i]);
}

// ---------------------------------------------------------------------------
// 4) xw = x @ W via V_WMMA_F32_16X16X4_F32 (exact f32 path).
//    Block = 256 threads = 8 waves. Block b handles rows [16b, 16b+16).
//    Wave w handles output columns [16w, 16w+16). K=128 -> 32 WMMAs.
//
//    A (16x4 f32) lane layout (ISA 7.12.2): lanes 0-15 hold M=lane, K={k0,k0+1};
//    lanes 16-31 hold M=lane-16, K={k0+2,k0+3}.  B (4x16) symmetric in N.
//    D (16x16 f32): VGPR r, lanes 0-15 -> M=r, lanes 16-31 -> M=r+8, N=lane%16.
// ---------------------------------------------------------------------------
__global__ __launch_bounds__(256) void gemm_wmma_kernel(
    const float* __restrict__ x, const float* __restrict__ W,
    float* __restrict__ xw, int n) {
    const int wave = threadIdx.x >> 5;
    const int lane = threadIdx.x & 31;
    const int half = lane >> 4;   // 0: K pair 0, 1: K pair 1
    const int l    = lane & 15;

    const int row = min(blockIdx.x * 16 + l, n - 1);  // A row fed by this lane
    const int col = wave * 16 + l;                    // B/D column

    const float* __restrict__ arow = x + (size_t)row * DIM;

    v8f acc = {};  // C starts at zero
#pragma unroll
    for (int k0 = 0; k0 < DIM; k0 += 4) {
        const int ka = k0 + (half << 1);
        v2f a = *(const v2f*)(arow + ka);            // global_load_b64
        v2f b;
        b.x = W[(size_t)ka * DIM + col];
        b.y = W[(size_t)(ka + 1) * DIM + col];
        acc = __builtin_amdgcn_wmma_f32_16x16x4_f32(
            /*neg_a=*/false, a, /*neg_b=*/false, b,
            /*c_mod=*/(short)0, acc, /*reuse_a=*/false, /*reuse_b=*/false);
    }

    const int m0 = blockIdx.x * 16 + half * 8;
#pragma unroll
    for (int r = 0; r < 8; ++r) {
        const int m = m0 + r;
        if (m < n) xw[(size_t)m * DIM + col] = acc[r];
    }
}

// ---------------------------------------------------------------------------
// 5) out[node] = xw[node] * dinv[node]^2   (self-loop message; also fully
//    initializes d_out each call). One thread per float4.
// ---------------------------------------------------------------------------
__global__ void selfloop_kernel(const float* __restrict__ xw,
                                const float* __restrict__ dinv,
                                float* __restrict__ out, int n) {
    int i = blockIdx.x * blockDim.x + threadIdx.x;  // over n * (DIM/4)
    if (i >= n * (DIM / 4)) return;
    const int node = i >> 5;  // DIM/4 == 32 float4 per node
    float s = dinv[node];
    s = s * s;
    float4 v = ((const float4*)xw)[i];
    float4 o = {v.x * s, v.y * s, v.z * s, v.w * s};
    ((float4*)out)[i] = o;
}

// ---------------------------------------------------------------------------
// 6) One wave per edge. norm = dinv[src]*dinv[dst]; each lane owns 4 columns:
//    float4 gather from xw[src] (L2-resident), 4x global_atomic_add_f32 into
//    out[dst] (L2-resident).
// ---------------------------------------------------------------------------
__global__ __launch_bounds__(256) void scatter_kernel(
    const long long* __restrict__ src, const long long* __restrict__ dst,
    const float* __restrict__ xw, const float* __restrict__ dinv,
    float* __restrict__ out, int e) {
    const int w    = (blockIdx.x * blockDim.x + threadIdx.x) >> 5;
    const int lane = threadIdx.x & 31;
    if (w >= e) return;
    const int s = (int)src[w];
    const int d = (int)dst[w];
    const float norm = dinv[s] * dinv[d];

    const float4 v = *(const float4*)(xw + (size_t)s * DIM + lane * 4);
    float* o = out + (size_t)d * DIM + lane * 4;
    atomicAdd(o + 0, v.x * norm);
    atomicAdd(o + 1, v.y * norm);
    atomicAdd(o + 2, v.z * norm);
    atomicAdd(o + 3, v.w * norm);
}

// ---------------------------------------------------------------------------
// 7) out = relu(out + bias), float4-vectorized.
// ---------------------------------------------------------------------------
__global__ void finalize_kernel(float* __restrict__ out,
                                const float* __restrict__ bias, int n) {
    int i = blockIdx.x * blockDim.x + threadIdx.x;  // over n * 32
    if (i >= n * (DIM / 4)) return;
    const float4 bb = ((const float4*)bias)[i & 31];
    float4 v = ((float4*)out)[i];
    v.x = fmaxf(v.x + bb.x, 0.0f);
    v.y = fmaxf(v.y + bb.y, 0.0f);
    v.z = fmaxf(v.z + bb.z, 0.0f);
    v.w = fmaxf(v.w + bb.w, 0.0f);
    ((float4*)out)[i] = v;
}

// ---------------------------------------------------------------------------
// Inputs (setup_inputs order): x[N*128] f32, edge_index[2*E] i64,
// W[128*128] f32, b[128] f32. Output: relu(GCNConv(x)) [N*128] f32.
// Workspace: dinv (N f32) | xw (N*128 f32)  -> ~51.6 MB.
// ---------------------------------------------------------------------------
extern "C" void kernel_launch(void* const* d_in, const int* in_sizes, int n_in,
                              void* d_out, int out_size, void* d_ws, size_t ws_size,
                              hipStream_t stream) {
    const float*     x  = (const float*)d_in[0];
    const long long* ei = (const long long*)d_in[1];
    const float*     W  = (const float*)d_in[2];
    const float*     b  = (const float*)d_in[3];
    float*           out = (float*)d_out;

    const int n = in_sizes[0] / DIM;   // 100000
    const int e = in_sizes[1] / 2;     // 1600000

    float* dinv = (float*)d_ws;        // n floats (deg, then deg^-1/2 in place)
    float* xw   = dinv + n;            // n*DIM floats

    const long long* srcp = ei;        // edge_index[0]
    const long long* dstp = ei + e;    // edge_index[1]

    deg_init_kernel<<<(n + 255) / 256, 256, 0, stream>>>(dinv, n);
    deg_accum_kernel<<<(e + 255) / 256, 256, 0, stream>>>(dstp, dinv, e);
    dinv_kernel<<<(n + 255) / 256, 256, 0, stream>>>(dinv, n);

    gemm_wmma_kernel<<<(n + 15) / 16, 256, 0, stream>>>(x, W, xw, n);

    const int nvec = n * (DIM / 4);
    selfloop_kernel<<<(nvec + 255) / 256, 256, 0, stream>>>(xw, dinv, out, n);

    scatter_kernel<<<(e + 7) / 8, 256, 0, stream>>>(srcp, dstp, xw, dinv, out, e);

    finalize_kernel<<<(nvec + 255) / 256, 256, 0, stream>>>(out, b, n);
}